// GroupAwareEncoder_76038101008443
// MI455X (gfx1250) — compile-verified
//
#include <hip/hip_runtime.h>

typedef __attribute__((ext_vector_type(2))) float v2f;
typedef __attribute__((ext_vector_type(8))) float v8f;

#define N_USERS 60000
#define N_ITEMS 40000
#define NTOT    100000
#define EMB     64
#define LN_EPS  1e-5f

// ---------------------------------------------------------------------------
// Kernel A: ego_t = par0*par1*(filt ⊙ X) @ W + X   (per user/item region)
// One wave computes a 16x64 output tile with V_WMMA_F32_16X16X4_F32.
// 4 waves per block (128 threads) -> 64 rows per block.
// ---------------------------------------------------------------------------
__global__ __launch_bounds__(128)
void k_dense(const float* __restrict__ ego,
             const float* __restrict__ w_uu, const float* __restrict__ filt_uu,
             const float* __restrict__ par_uu,
             const float* __restrict__ w_ii, const float* __restrict__ filt_ii,
             const float* __restrict__ par_ii,
             float* __restrict__ ego_t)
{
    __shared__ float Wu[64 * 64];
    __shared__ float Wi[64 * 64];
    __shared__ float Xs[4][16 * 64];

    const int tid = threadIdx.x;

    // Cooperative load of both 64x64 weight matrices (float4, coalesced).
    for (int i = tid; i < 64 * 64 / 4; i += 128) {
        ((float4*)Wu)[i] = ((const float4*)w_uu)[i];
        ((float4*)Wi)[i] = ((const float4*)w_ii)[i];
    }

    const int wave = tid >> 5;
    const int lane = tid & 31;
    const int r0 = blockIdx.x * 64 + wave * 16;   // first row of this wave's tile

    // Stage the raw 16x64 X tile into LDS (float4, coalesced). OOB rows -> 0.
    for (int i = 0; i < 8; ++i) {
        int flat = i * 128 + lane * 4;            // element index within tile
        int row = r0 + (flat >> 6);
        float4 v = make_float4(0.f, 0.f, 0.f, 0.f);
        if (row < NTOT) v = ((const float4*)(ego + (size_t)row * EMB))[(flat & 63) >> 2];
        ((float4*)Xs[wave])[flat >> 2] = v;
    }
    __syncthreads();

    // Wave-uniform region selection (tiles are 16-aligned; 60000 % 16 == 0).
    const bool isUser = (r0 < N_USERS);
    const float* __restrict__ Wl = isUser ? Wu : Wi;
    const float s = isUser ? (par_uu[0] * par_uu[1]) : (par_ii[0] * par_ii[1]);

    // A-matrix layout (16x4 f32): lane L holds row m = L%16, K = (L/16)*2 + j.
    const int m = lane & 15;
    const int khalf = lane >> 4;
    const int rowA = r0 + m;
    float f = 0.f;
    if (rowA < NTOT)
        f = s * (isUser ? filt_uu[rowA] : filt_ii[rowA - N_USERS]);

    v8f acc0 = {}, acc1 = {}, acc2 = {}, acc3 = {};
    const float* __restrict__ xrow = &Xs[wave][m * 64];

    for (int kk = 0; kk < 64; kk += 4) {
        const int k = kk + khalf * 2;
        v2f a;
        a.x = f * xrow[k];
        a.y = f * xrow[k + 1];

        v2f b0, b1, b2, b3;                        // B layout mirrors A (n = L%16)
        b0.x = Wl[k * 64 + 0  + m]; b0.y = Wl[(k + 1) * 64 + 0  + m];
        b1.x = Wl[k * 64 + 16 + m]; b1.y = Wl[(k + 1) * 64 + 16 + m];
        b2.x = Wl[k * 64 + 32 + m]; b2.y = Wl[(k + 1) * 64 + 32 + m];
        b3.x = Wl[k * 64 + 48 + m]; b3.y = Wl[(k + 1) * 64 + 48 + m];

        acc0 = __builtin_amdgcn_wmma_f32_16x16x4_f32(false, a, false, b0, (short)0, acc0, false, false);
        acc1 = __builtin_amdgcn_wmma_f32_16x16x4_f32(false, a, false, b1, (short)0, acc1, false, false);
        acc2 = __builtin_amdgcn_wmma_f32_16x16x4_f32(false, a, false, b2, (short)0, acc2, false, false);
        acc3 = __builtin_amdgcn_wmma_f32_16x16x4_f32(false, a, false, b3, (short)0, acc3, false, false);
    }

    // D layout: VGPR v holds D[v + (lane/16)*8][lane%16]. Fuse residual +X.
    for (int v = 0; v < 8; ++v) {
        const int rloc = v + khalf * 8;
        const int rowD = r0 + rloc;
        if (rowD < NTOT) {
            float* __restrict__ orow = ego_t + (size_t)rowD * EMB;
            const float* __restrict__ xr = &Xs[wave][rloc * 64];
            orow[0  + m] = acc0[v] + xr[0  + m];
            orow[16 + m] = acc1[v] + xr[16 + m];
            orow[32 + m] = acc2[v] + xr[32 + m];
            orow[48 + m] = acc3[v] + xr[48 + m];
        }
    }
}

// ---------------------------------------------------------------------------
// Zero a buffer (workspace is poisoned, must self-initialize every call).
// ---------------------------------------------------------------------------
__global__ __launch_bounds__(256)
void k_zero(float4* __restrict__ p, long n4)
{
    long i = (long)blockIdx.x * blockDim.x + threadIdx.x;
    long stride = (long)gridDim.x * blockDim.x;
    float4 z = make_float4(0.f, 0.f, 0.f, 0.f);
    for (; i < n4; i += stride) p[i] = z;
}

// ---------------------------------------------------------------------------
// Scatter SpMM: out[dst[e]] += vals[e] * x[src[e]].
// 16 lanes per edge, float4 per lane -> 4 global_atomic_add_f32 each.
// Accumulators (25.6 MB) live in the 192 MB L2, so atomics stay on-chip.
// ---------------------------------------------------------------------------
__global__ __launch_bounds__(256)
void k_scatter(const int* __restrict__ src_idx, const int* __restrict__ dst_idx,
               const float* __restrict__ vals, const float* __restrict__ x,
               float* __restrict__ out, int nnz)
{
    const int t = blockIdx.x * 256 + threadIdx.x;
    const int e = t >> 4;
    const int sub = t & 15;
    if (e >= nnz) return;

    const int r = src_idx[e];
    const int c = dst_idx[e];
    const float v = vals[e];

    const float4 xv = ((const float4*)(x + (size_t)r * EMB))[sub];
    float* __restrict__ o = out + (size_t)c * EMB + sub * 4;
    unsafeAtomicAdd(o + 0, v * xv.x);
    unsafeAtomicAdd(o + 1, v * xv.y);
    unsafeAtomicAdd(o + 2, v * xv.z);
    unsafeAtomicAdd(o + 3, v * xv.w);
}

// ---------------------------------------------------------------------------
// LayerNorm + residual: one row per wave32, 2 elements per lane,
// wave reduction via __shfl_xor.
// ---------------------------------------------------------------------------
__global__ __launch_bounds__(256)
void k_ln(const float* __restrict__ y, const float* __restrict__ ego,
          const float* __restrict__ gamma, const float* __restrict__ beta,
          float* __restrict__ out)
{
    const int wave = threadIdx.x >> 5;
    const int lane = threadIdx.x & 31;
    const int row = blockIdx.x * 8 + wave;
    if (row >= NTOT) return;                       // wave-uniform exit

    const float2 v = ((const float2*)(y + (size_t)row * EMB))[lane];
    float sum = v.x + v.y;
    float sq  = v.x * v.x + v.y * v.y;
    for (int off = 16; off > 0; off >>= 1) {
        sum += __shfl_xor(sum, off, 32);
        sq  += __shfl_xor(sq,  off, 32);
    }
    const float mu  = sum * (1.f / 64.f);
    const float var = sq * (1.f / 64.f) - mu * mu;
    const float inv = rsqrtf(var + LN_EPS);

    const int c = lane * 2;
    const float2 r = ((const float2*)(ego + (size_t)row * EMB))[lane];
    float2 o;
    o.x = (v.x - mu) * inv * gamma[c]     + beta[c]     + r.x;
    o.y = (v.y - mu) * inv * gamma[c + 1] + beta[c + 1] + r.y;
    ((float2*)(out + (size_t)row * EMB))[lane] = o;
}

// ---------------------------------------------------------------------------
extern "C" void kernel_launch(void* const* d_in, const int* in_sizes, int n_in,
                              void* d_out, int out_size, void* d_ws, size_t ws_size,
                              hipStream_t stream)
{
    const float* ego     = (const float*)d_in[0];
    const int*   rows    = (const int*)  d_in[1];
    const int*   cols    = (const int*)  d_in[2];
    const float* vals    = (const float*)d_in[3];
    const float* w_uu    = (const float*)d_in[4];
    const float* filt_uu = (const float*)d_in[5];
    const float* par_uu  = (const float*)d_in[6];
    const float* w_ii    = (const float*)d_in[7];
    const float* filt_ii = (const float*)d_in[8];
    const float* par_ii  = (const float*)d_in[9];
    const float* gamma   = (const float*)d_in[10];
    const float* beta    = (const float*)d_in[11];
    float* out = (float*)d_out;
    const int nnz = in_sizes[1];

    // Workspace: buf0 = ego_t (later reused as y), buf1 = h. 51.2 MB total.
    float* buf0 = (float*)d_ws;
    float* buf1 = buf0 + (size_t)NTOT * EMB;
    const long n4 = (long)NTOT * EMB / 4;
    const int zgrid = 2048;

    // 1. zero h
    k_zero<<<zgrid, 256, 0, stream>>>((float4*)buf1, n4);
    // 2. dense WMMA transform -> ego_t (buf0)
    k_dense<<<(NTOT + 63) / 64, 128, 0, stream>>>(ego, w_uu, filt_uu, par_uu,
                                                  w_ii, filt_ii, par_ii, buf0);
    // 3. h[cols[e]] += vals[e] * ego_t[rows[e]]
    {
        long threads = (long)nnz * 16;
        int blocks = (int)((threads + 255) / 256);
        k_scatter<<<blocks, 256, 0, stream>>>(rows, cols, vals, buf0, buf1, nnz);
    }
    // 4. zero y (reuse buf0 — ego_t no longer needed)
    k_zero<<<zgrid, 256, 0, stream>>>((float4*)buf0, n4);
    // 5. y[rows[e]] += vals[e] * h[cols[e]]
    {
        long threads = (long)nnz * 16;
        int blocks = (int)((threads + 255) / 256);
        k_scatter<<<blocks, 256, 0, stream>>>(cols, rows, vals, buf1, buf0, nnz);
    }
    // 6. LayerNorm + residual -> out
    k_ln<<<(NTOT + 7) / 8, 256, 0, stream>>>(buf0, ego, gamma, beta, out);
}